// CGNet_20684562497950
// MI455X (gfx1250) — compile-verified
//
#include <hip/hip_runtime.h>
#include <hip/hip_bf16.h>

typedef __attribute__((ext_vector_type(2))) float v2f;
typedef __attribute__((ext_vector_type(8))) float v8f;

#define HID 64
#define NUM_ATOMS 26
#define FEATD 90   // EMB + NUM_ATOMS
#define NITER 3

// ---------------------------------------------------------------------------
// f32 WMMA helpers (V_WMMA_F32_16X16X4_F32, wave32)
// A 16x4: lanes 0-15 hold {K,K+1} of row M=lane; lanes 16-31 hold {K+2,K+3} of M=lane-16
// B 4x16: lanes 0-15 hold {K,K+1} of col N=lane; lanes 16-31 hold {K+2,K+3} of N=lane-16
// C 16x16: lanes 0-15 -> rows 0-7 (col=lane), lanes 16-31 -> rows 8-15 (col=lane-16)
// ---------------------------------------------------------------------------
__device__ __forceinline__ v8f wmma4(v2f a, v2f b, v8f c) {
  return __builtin_amdgcn_wmma_f32_16x16x4_f32(false, a, false, b, (short)0, c,
                                               false, false);
}

// One wave accumulates a 16x16 f32 tile: C[0:16, n0:n0+16] += A(16xK) * B(KxN)
// A is row-major with leading dim lda (rows 0..15 must be readable; zero-pad).
// B: bT=false -> B[k*ldb + n] ; bT=true -> B[n*ldb + k] (i.e. logical B = Bmem^T)
__device__ __forceinline__ v8f wmma_gemm16(const float* A, int lda,
                                           const float* B, int ldb, bool bT,
                                           int n0, int K, v8f acc) {
  const int lane = threadIdx.x & 31;
  const bool hi = lane >= 16;
  const int l = lane & 15;
  const int n = n0 + l;
  for (int k = 0; k < K; k += 4) {
    const int ka = hi ? (k + 2) : k;
    v2f a; a.x = A[l * lda + ka]; a.y = A[l * lda + ka + 1];
    v2f b;
    if (bT) { b.x = B[(size_t)n * ldb + ka];  b.y = B[(size_t)n * ldb + ka + 1]; }
    else    { b.x = B[(size_t)ka * ldb + n];  b.y = B[(size_t)(ka + 1) * ldb + n]; }
    acc = wmma4(a, b, acc);
  }
  return acc;
}

// ---------------------------------------------------------------------------
// out0 = relu(x @ W_in + b_in)        x:(N,26)  W_in:(26,64)
// ---------------------------------------------------------------------------
__global__ void input_mlp_kernel(const float* __restrict__ x,
                                 const float* __restrict__ Win,
                                 const float* __restrict__ bin,
                                 float* __restrict__ hbuf, int N) {
  int n = blockIdx.x * 4 + (threadIdx.x >> 6);
  int j = threadIdx.x & 63;
  if (n >= N) return;
  const float* xr = x + (size_t)n * NUM_ATOMS;
  float acc = bin[j];
  #pragma unroll
  for (int i = 0; i < NUM_ATOMS; ++i) acc += xr[i] * Win[i * HID + j];
  hbuf[(size_t)n * HID + j] = fmaxf(acc, 0.f);
}

// ---------------------------------------------------------------------------
// h1 = relu(edge_attr @ We1 + be1)    edge_attr:(E,4)  We1:(4,128)
// ---------------------------------------------------------------------------
__global__ void edge_mlp1_kernel(const float* __restrict__ ea,
                                 const float* __restrict__ We1,
                                 const float* __restrict__ be1,
                                 float* __restrict__ h1, int E) {
  int e = blockIdx.x * 2 + (threadIdx.x >> 7);
  int j = threadIdx.x & 127;
  if (e >= E) return;
  const float* a = ea + (size_t)e * 4;
  float acc = be1[j];
  #pragma unroll
  for (int i = 0; i < 4; ++i) acc += a[i] * We1[i * 128 + j];
  h1[(size_t)e * 128 + j] = fmaxf(acc, 0.f);
}

// ---------------------------------------------------------------------------
// Fused NNConv message GEMM:
//   msg[e,o] = sum_{k,i} h1[e,k]*h[src_e,i]*We2[k, i*64+o]  +  sum_i h[src_e,i]*be2[i*64+o]
// = (Q @ We2_view), Q built on the fly in registers; K = 8192 (+64 bias steps).
// Block: 128 threads (4 waves), 64 edges. Wave w owns output cols [16w,16w+16);
// 4 edge sub-tiles of 16 -> B fragment reused 4x per K-chunk.
// ---------------------------------------------------------------------------
__global__ void __launch_bounds__(128)
msg_wmma_kernel(const float* __restrict__ h, const int* __restrict__ src,
                const float* __restrict__ h1, const float* __restrict__ We2,
                const float* __restrict__ be2, float* __restrict__ msg, int E) {
  __shared__ float s_s[64][68];    // gathered h[src] (padded: bank = 4e + i)
  __shared__ float s_h1[64][132];  // h1 tile        (padded: bank = 4e + k)
  const int eb = blockIdx.x * 64;
  const int tid = threadIdx.x;

  for (int idx = tid; idx < 64 * 64; idx += 128) {
    int e = idx >> 6, j = idx & 63;
    int ge = eb + e;
    float v = 0.f;
    if (ge < E) v = h[(size_t)src[ge] * HID + j];
    s_s[e][j] = v;
  }
  for (int idx = tid; idx < 64 * 128; idx += 128) {
    int e = idx >> 7, j = idx & 127;
    int ge = eb + e;
    s_h1[e][j] = (ge < E) ? h1[(size_t)ge * 128 + j] : 0.f;
  }
  __syncthreads();

  const int w = tid >> 5;
  const int lane = tid & 31;
  const bool hi = lane >= 16;
  const int l = lane & 15;
  const int n0 = w * 16;
  const int toff = hi ? 2 : 0;

  v8f acc[4] = {{}, {}, {}, {}};

  // main K loop: 2048 chunks of 4 over flattened (k,i)
  #pragma unroll 4
  for (int c = 0; c < 2048; ++c) {
    const int k = c >> 4;
    const int i0 = (c & 15) << 2;
    const float* Bp = We2 + (size_t)k * 4096 + (size_t)(i0 + toff) * 64 + n0 + l;
    v2f b; b.x = Bp[0]; b.y = Bp[64];
    #pragma unroll
    for (int et = 0; et < 4; ++et) {
      const int e = et * 16 + l;
      const float hv = s_h1[e][k];
      v2f a; a.x = hv * s_s[e][i0 + toff]; a.y = hv * s_s[e][i0 + toff + 1];
      acc[et] = wmma4(a, b, acc[et]);
    }
  }
  // be2 bias contribution: msg += s @ reshape(be2, 64, 64)
  #pragma unroll 4
  for (int c = 0; c < 16; ++c) {
    const int i0 = c << 2;
    const float* Bp = be2 + (size_t)(i0 + toff) * 64 + n0 + l;
    v2f b; b.x = Bp[0]; b.y = Bp[64];
    #pragma unroll
    for (int et = 0; et < 4; ++et) {
      const int e = et * 16 + l;
      v2f a; a.x = s_s[e][i0 + toff]; a.y = s_s[e][i0 + toff + 1];
      acc[et] = wmma4(a, b, acc[et]);
    }
  }

  const int rbase = hi ? 8 : 0;
  #pragma unroll
  for (int et = 0; et < 4; ++et)
    #pragma unroll
    for (int r = 0; r < 8; ++r) {
      const int ge = eb + et * 16 + rbase + r;
      if (ge < E) msg[(size_t)ge * HID + n0 + l] = acc[et][r];
    }
}

// ---------------------------------------------------------------------------
// agg = segment_sum(msg, dst) via native f32 global atomics
// ---------------------------------------------------------------------------
__global__ void scatter_add_kernel(const float* __restrict__ msg,
                                   const int* __restrict__ dst,
                                   float* __restrict__ agg, int E) {
  int idx = blockIdx.x * 256 + threadIdx.x;
  if (idx >= E * HID) return;
  int e = idx >> 6;
  unsafeAtomicAdd(&agg[(size_t)dst[e] * HID + (idx & 63)], msg[idx]);
}

// ---------------------------------------------------------------------------
// Fused conv-root + GRU cell for a tile of 16 nodes (block = 128 thr = 4 waves)
//   m  = relu(h @ W_root + agg + b_conv)
//   gi = m @ W_ih^T + b_ih ; gh = h @ W_hh^T + b_hh ; gates ; h <- new h
// ---------------------------------------------------------------------------
__global__ void __launch_bounds__(128)
node_update_kernel(float* __restrict__ hbuf, const float* __restrict__ agg,
                   const float* __restrict__ Wroot, const float* __restrict__ bconv,
                   const float* __restrict__ Wih, const float* __restrict__ bih,
                   const float* __restrict__ Whh, const float* __restrict__ bhh,
                   int N) {
  __shared__ float s_h[16][68];
  __shared__ float s_m[16][68];
  __shared__ float s_gi[16][196];
  __shared__ float s_gh[16][196];
  const int nt = blockIdx.x * 16;
  const int tid = threadIdx.x;

  for (int idx = tid; idx < 16 * 64; idx += 128) {
    int r = idx >> 6, j = idx & 63;
    int n = nt + r;
    s_h[r][j] = (n < N) ? hbuf[(size_t)n * HID + j] : 0.f;
  }
  __syncthreads();

  const int w = tid >> 5;
  const int lane = tid & 31;
  const bool hi = lane >= 16;
  const int l = lane & 15;
  const int rbase = hi ? 8 : 0;

  { // m = relu(h @ W_root + agg + b_conv), wave w -> cols [16w, 16w+16)
    const int n0 = w * 16;
    v8f a = {};
    a = wmma_gemm16(&s_h[0][0], 68, Wroot, HID, false, n0, HID, a);
    #pragma unroll
    for (int r = 0; r < 8; ++r) {
      int row = rbase + r, n = nt + row, col = n0 + l;
      float v = a[r] + bconv[col] + ((n < N) ? agg[(size_t)n * HID + col] : 0.f);
      s_m[row][col] = fmaxf(v, 0.f);
    }
  }
  __syncthreads();

  // gi / gh: 12 column tiles of 16 over 192; wave w does tiles w, w+4, w+8
  for (int t = w; t < 12; t += 4) {
    const int n0 = t * 16;
    v8f gi = {}, gh = {};
    gi = wmma_gemm16(&s_m[0][0], 68, Wih, HID, true, n0, HID, gi);
    gh = wmma_gemm16(&s_h[0][0], 68, Whh, HID, true, n0, HID, gh);
    #pragma unroll
    for (int r = 0; r < 8; ++r) {
      int row = rbase + r, col = n0 + l;
      s_gi[row][col] = gi[r] + bih[col];
      s_gh[row][col] = gh[r] + bhh[col];
    }
  }
  __syncthreads();

  for (int idx = tid; idx < 16 * 64; idx += 128) {
    int row = idx >> 6, col = idx & 63;
    int n = nt + row;
    if (n >= N) continue;
    float rg = 1.f / (1.f + expf(-(s_gi[row][col] + s_gh[row][col])));
    float zg = 1.f / (1.f + expf(-(s_gi[row][64 + col] + s_gh[row][64 + col])));
    float ng = tanhf(s_gi[row][128 + col] + rg * s_gh[row][128 + col]);
    hbuf[(size_t)n * HID + col] = (1.f - zg) * ng + zg * s_h[row][col];
  }
}

// ---------------------------------------------------------------------------
// o2 = relu(h @ Wo1 + bo1) @ Wo2 + bo2
// ---------------------------------------------------------------------------
__global__ void __launch_bounds__(128)
out_mlp_kernel(const float* __restrict__ hbuf,
               const float* __restrict__ Wo1, const float* __restrict__ bo1,
               const float* __restrict__ Wo2, const float* __restrict__ bo2,
               float* __restrict__ o2g, int N) {
  __shared__ float s_h[16][68];
  __shared__ float s_o1[16][68];
  const int nt = blockIdx.x * 16;
  const int tid = threadIdx.x;
  for (int idx = tid; idx < 16 * 64; idx += 128) {
    int r = idx >> 6, j = idx & 63;
    int n = nt + r;
    s_h[r][j] = (n < N) ? hbuf[(size_t)n * HID + j] : 0.f;
  }
  __syncthreads();
  const int w = tid >> 5, lane = tid & 31;
  const bool hi = lane >= 16;
  const int l = lane & 15, n0 = w * 16, rbase = hi ? 8 : 0;

  v8f a1 = {};
  a1 = wmma_gemm16(&s_h[0][0], 68, Wo1, HID, false, n0, HID, a1);
  #pragma unroll
  for (int r = 0; r < 8; ++r)
    s_o1[rbase + r][n0 + l] = fmaxf(a1[r] + bo1[n0 + l], 0.f);
  __syncthreads();

  v8f a2 = {};
  a2 = wmma_gemm16(&s_o1[0][0], 68, Wo2, HID, false, n0, HID, a2);
  #pragma unroll
  for (int r = 0; r < 8; ++r) {
    int n = nt + rbase + r;
    if (n < N) o2g[(size_t)n * HID + n0 + l] = a2[r] + bo2[n0 + l];
  }
}

// ---------------------------------------------------------------------------
// feat = normalize([o2, x]); also accumulate per-graph sums + counts
// ---------------------------------------------------------------------------
__global__ void __launch_bounds__(128)
feat_kernel(const float* __restrict__ o2, const float* __restrict__ x,
            const int* __restrict__ batch, float* __restrict__ feat_out,
            float* __restrict__ gsum, int N) {
  const int n = blockIdx.x;
  if (n >= N) return;
  __shared__ float sv[FEATD];
  __shared__ float red[4];
  const int tid = threadIdx.x;
  float v = 0.f;
  if (tid < HID)        { v = o2[(size_t)n * HID + tid]; sv[tid] = v; }
  else if (tid < FEATD) { v = x[(size_t)n * NUM_ATOMS + (tid - HID)]; sv[tid] = v; }
  float sq = v * v;
  #pragma unroll
  for (int off = 16; off > 0; off >>= 1) sq += __shfl_down(sq, off, 32);
  if ((tid & 31) == 0) red[tid >> 5] = sq;
  __syncthreads();
  const float inv = 1.f / fmaxf(sqrtf(red[0] + red[1] + red[2] + red[3]), 1e-12f);
  if (tid < FEATD) {
    float fv = sv[tid] * inv;
    feat_out[(size_t)n * FEATD + tid] = fv;
    unsafeAtomicAdd(&gsum[(size_t)batch[n] * (FEATD + 1) + tid], fv);
  }
  if (tid == FEATD) unsafeAtomicAdd(&gsum[(size_t)batch[n] * (FEATD + 1) + FEATD], 1.f);
}

// ---------------------------------------------------------------------------
// ratio = sigmoid((gsum/cnt) @ Wp + bp)
// ---------------------------------------------------------------------------
__global__ void ratio_kernel(const float* __restrict__ gsum,
                             const float* __restrict__ Wp,
                             const float* __restrict__ bp,
                             float* __restrict__ out, int G) {
  int g = blockIdx.x * 64 + threadIdx.x;
  if (g >= G) return;
  float cnt = fmaxf(gsum[(size_t)g * (FEATD + 1) + FEATD], 1.f);
  float acc = bp[0];
  for (int j = 0; j < FEATD; ++j)
    acc += (gsum[(size_t)g * (FEATD + 1) + j] / cnt) * Wp[j];
  out[g] = 1.f / (1.f + expf(-acc));
}

// ---------------------------------------------------------------------------
extern "C" void kernel_launch(void* const* d_in, const int* in_sizes, int n_in,
                              void* d_out, int out_size, void* d_ws, size_t ws_size,
                              hipStream_t stream) {
  const float* x      = (const float*)d_in[0];
  const int*   eidx   = (const int*)  d_in[1];
  const float* eattr  = (const float*)d_in[2];
  const int*   batch  = (const int*)  d_in[3];
  const float* Win    = (const float*)d_in[4];
  const float* bin    = (const float*)d_in[5];
  const float* We1    = (const float*)d_in[6];
  const float* be1    = (const float*)d_in[7];
  const float* We2    = (const float*)d_in[8];
  const float* be2    = (const float*)d_in[9];
  const float* Wroot  = (const float*)d_in[10];
  const float* bconv  = (const float*)d_in[11];
  const float* Wih    = (const float*)d_in[12];
  const float* bih    = (const float*)d_in[13];
  const float* Whh    = (const float*)d_in[14];
  const float* bhh    = (const float*)d_in[15];
  const float* Wo1    = (const float*)d_in[16];
  const float* bo1    = (const float*)d_in[17];
  const float* Wo2    = (const float*)d_in[18];
  const float* bo2    = (const float*)d_in[19];
  const float* Wp     = (const float*)d_in[20];
  const float* bp     = (const float*)d_in[21];
  (void)n_in; (void)ws_size;

  const int N = in_sizes[0] / NUM_ATOMS;      // 25000
  const int E = in_sizes[1] / 2;              // 50000
  const int G = out_size - N * FEATD;         // 64 graphs
  const int* src = eidx;
  const int* dst = eidx + E;

  float* ws   = (float*)d_ws;
  float* hbuf = ws;                 size_t off = (size_t)N * HID;
  float* h1   = ws + off;           off += (size_t)E * 128;
  float* msg  = ws + off;           off += (size_t)E * HID;
  float* agg  = ws + off;           off += (size_t)N * HID;
  float* o2   = ws + off;           off += (size_t)N * HID;
  float* gsum = ws + off;           off += (size_t)G * (FEATD + 1);

  float* feat_out  = (float*)d_out;
  float* ratio_out = feat_out + (size_t)N * FEATD;

  // out0 = relu(x @ W_in + b_in); h1 = relu(edge_attr @ We1 + be1)
  input_mlp_kernel<<<dim3((N + 3) / 4), dim3(256), 0, stream>>>(x, Win, bin, hbuf, N);
  edge_mlp1_kernel<<<dim3((E + 1) / 2), dim3(256), 0, stream>>>(eattr, We1, be1, h1, E);

  const int msg_blocks  = (E + 63) / 64;
  const int scat_blocks = (E * HID + 255) / 256;
  const int node_blocks = (N + 15) / 16;
  for (int it = 0; it < NITER; ++it) {
    msg_wmma_kernel<<<dim3(msg_blocks), dim3(128), 0, stream>>>(
        hbuf, src, h1, We2, be2, msg, E);
    hipMemsetAsync(agg, 0, (size_t)N * HID * sizeof(float), stream);
    scatter_add_kernel<<<dim3(scat_blocks), dim3(256), 0, stream>>>(msg, dst, agg, E);
    node_update_kernel<<<dim3(node_blocks), dim3(128), 0, stream>>>(
        hbuf, agg, Wroot, bconv, Wih, bih, Whh, bhh, N);
  }

  out_mlp_kernel<<<dim3(node_blocks), dim3(128), 0, stream>>>(
      hbuf, Wo1, bo1, Wo2, bo2, o2, N);
  hipMemsetAsync(gsum, 0, (size_t)G * (FEATD + 1) * sizeof(float), stream);
  feat_kernel<<<dim3(N), dim3(128), 0, stream>>>(o2, x, batch, feat_out, gsum, N);
  ratio_kernel<<<dim3((G + 63) / 64), dim3(64), 0, stream>>>(gsum, Wp, bp, ratio_out, G);
}